// GaussianPFR_19954418057864
// MI455X (gfx1250) — compile-verified
//
#include <hip/hip_runtime.h>
#include <math.h>

// ---------------------------------------------------------------------------
// GaussianPFR 'det' encode for MI455X (gfx1250, wave32).
//
// Algebra: argmax_s Σ_d [ p²(½-½/σ²) + p·(μ/σ²) ]  ==  row-argmax of the
// GEMM  score = W (bs×32) · F^T (32×n),  W=[μ/σ² | ½-½/σ²], F=[p | p²].
// Native fp32 WMMA (v_wmma_f32_16x16x4_f32), K=32 in 8 steps; two sample
// tiles in flight per wave to keep two independent WMMA chains in the pipe.
// Assumes n_samples % 32 == 0 (harness: 4096).
// ---------------------------------------------------------------------------

typedef __attribute__((ext_vector_type(2))) float v2f;
typedef __attribute__((ext_vector_type(4))) float v4f;
typedef __attribute__((ext_vector_type(8))) float v8f;

#define DIM    16
#define WAVES  8      // waves per block (8 * wave32 = 256 threads)
#define BROWS  16     // batch rows per block (one 16-row WMMA M tile)

__device__ __forceinline__ void load_row16(const float* __restrict__ base,
                                           int s, float p[DIM]) {
    const v4f* pp = (const v4f*)(base + (size_t)s * DIM);
#pragma unroll
    for (int q = 0; q < 4; ++q) {
        v4f v = pp[q];
#pragma unroll
        for (int j = 0; j < 4; ++j) p[4 * q + j] = v[j];
    }
}

// Build the K-step-k B fragment from a 16-float sample row.
// ISA f32 4x16 B layout (mirror of A): lanes 0-15 hold K=Kb,Kb+1;
// lanes 16-31 hold K=Kb+2,Kb+3.  K<16 -> p, K>=16 -> p^2.
template <int K>
__device__ __forceinline__ v2f bfrag(const float p[DIM], int half) {
    const int Kb = 4 * K;
    float b0, b1;
    if (K < 4) {
        b0 = half ? p[Kb + 2] : p[Kb + 0];
        b1 = half ? p[Kb + 3] : p[Kb + 1];
    } else {
        const int d = Kb - 16;
        float s0 = half ? p[d + 2] : p[d + 0];
        float s1 = half ? p[d + 3] : p[d + 1];
        b0 = s0 * s0; b1 = s1 * s1;
    }
    v2f b; b[0] = b0; b[1] = b1;
    return b;
}

__global__ __launch_bounds__(256) void gpfr_score_argmax_kernel(
    const float* __restrict__ mu_q,        // [bs,16]
    const float* __restrict__ std_q,       // [bs,16]
    const float* __restrict__ prior,       // [n_samples,16]
    float*       __restrict__ out_samples, // [bs,16]
    int*         __restrict__ out_idx,     // [bs]
    int n_samples)
{
    __shared__ float lds_val[WAVES][32][8];
    __shared__ int   lds_idx[WAVES][32][8];

    const int tid  = threadIdx.x;
    const int wave = tid >> 5;
    const int lane = tid & 31;
    const int half = lane >> 4;     // 0 -> lanes 0..15, 1 -> lanes 16..31
    const int l16  = lane & 15;
    const int mrow = blockIdx.x * BROWS + l16;   // batch row for A-fragment

    // ---- A fragment: W[16 rows x K=32], loop-invariant over sample tiles ---
    float wlin[DIM], wquad[DIM];
#pragma unroll
    for (int d = 0; d < DIM; ++d) {
        float m = mu_q [mrow * DIM + d];
        float s = std_q[mrow * DIM + d];
        float inv_var = 1.0f / (s * s);
        wlin [d] = m * inv_var;             // coefficient of p
        wquad[d] = 0.5f - 0.5f * inv_var;   // coefficient of p^2
    }
    v2f aF[8];
#pragma unroll
    for (int k = 0; k < 8; ++k) {
        const int Kb = 4 * k;
        float a0, a1;
        if (k < 4) {                        // K in [0,16): linear block
            a0 = half ? wlin[Kb + 2] : wlin[Kb + 0];
            a1 = half ? wlin[Kb + 3] : wlin[Kb + 1];
        } else {                            // K in [16,32): quadratic block
            const int d = Kb - 16;
            a0 = half ? wquad[d + 2] : wquad[d + 0];
            a1 = half ? wquad[d + 3] : wquad[d + 1];
        }
        aF[k][0] = a0; aF[k][1] = a1;
    }

    // ---- scan sample tiles, two per iteration (two WMMA chains) -----------
    float best[8];
    int   bidx[8];
#pragma unroll
    for (int r = 0; r < 8; ++r) { best[r] = -INFINITY; bidx[r] = 0x7fffffff; }

    const int npairs = n_samples >> 5;      // 32 samples per iteration
    const int pf_clamp = n_samples - DIM;   // last valid row start
    for (int pr = wave; pr < npairs; pr += WAVES) {
        const int s0 = pr * 32 + l16;       // tile 0 sample owned by this lane
        const int s1 = s0 + 16;             // tile 1 sample

        // branchless prefetch of next pair's rows (clamped; always executed)
        {
            int spf = s0 + WAVES * 32;
            spf = spf > pf_clamp ? pf_clamp : spf;
            __builtin_prefetch(prior + (size_t)spf * DIM, 0, 1);
        }

        float p0[DIM], p1[DIM];
        load_row16(prior, s0, p0);
        load_row16(prior, s1, p1);

        v8f c0 = {}, c1 = {};
#pragma unroll
        for (int k = 0; k < 8; ++k) {
            v2f b0, b1;
            switch (k) {   // compile-time K via template (full unroll)
            case 0: b0 = bfrag<0>(p0, half); b1 = bfrag<0>(p1, half); break;
            case 1: b0 = bfrag<1>(p0, half); b1 = bfrag<1>(p1, half); break;
            case 2: b0 = bfrag<2>(p0, half); b1 = bfrag<2>(p1, half); break;
            case 3: b0 = bfrag<3>(p0, half); b1 = bfrag<3>(p1, half); break;
            case 4: b0 = bfrag<4>(p0, half); b1 = bfrag<4>(p1, half); break;
            case 5: b0 = bfrag<5>(p0, half); b1 = bfrag<5>(p1, half); break;
            case 6: b0 = bfrag<6>(p0, half); b1 = bfrag<6>(p1, half); break;
            default:b0 = bfrag<7>(p0, half); b1 = bfrag<7>(p1, half); break;
            }
            // two independent accumulate chains -> XDL pipeline overlap
            c0 = __builtin_amdgcn_wmma_f32_16x16x4_f32(
                     false, aF[k], false, b0, (short)0, c0, false, false);
            c1 = __builtin_amdgcn_wmma_f32_16x16x4_f32(
                     false, aF[k], false, b1, (short)0, c1, false, false);
        }

        // C layout: lanes 0-15 -> (M=r, N=l16); lanes 16-31 -> (M=8+r, N=l16).
        // Branchless running argmax; tile0 before tile1 keeps index order,
        // strict '>' keeps first occurrence within this lane.
#pragma unroll
        for (int r = 0; r < 8; ++r) {
            float v0  = c0[r];
            bool  g0  = v0 > best[r];
            best[r] = g0 ? v0 : best[r];
            bidx[r] = g0 ? s0 : bidx[r];
            float v1  = c1[r];
            bool  g1  = v1 > best[r];
            best[r] = g1 ? v1 : best[r];
            bidx[r] = g1 ? s1 : bidx[r];
        }
    }

    // ---- cross-lane / cross-wave reduction via LDS -------------------------
#pragma unroll
    for (int r = 0; r < 8; ++r) {
        lds_val[wave][lane][r] = best[r];
        lds_idx[wave][lane][r] = bidx[r];
    }
    __syncthreads();

    if (tid < BROWS) {
        const int row  = tid;
        const int h    = row >> 3;          // which lane half owns this M row
        const int slot = row & 7;           // accumulator VGPR slot
        float bv = -INFINITY;
        int   bi = 0x7fffffff;
#pragma unroll 1
        for (int w = 0; w < WAVES; ++w) {
#pragma unroll 1
            for (int l = 0; l < 16; ++l) {
                float v = lds_val[w][h * 16 + l][slot];
                int   i = lds_idx[w][h * 16 + l][slot];
                // emulate jnp.argmax first-max tie-breaking
                if (v > bv || (v == bv && i < bi)) { bv = v; bi = i; }
            }
        }
        const int ob = blockIdx.x * BROWS + row;
        out_idx[ob] = bi;
        const v4f* src = (const v4f*)(prior + (size_t)bi * DIM);
        v4f*       dst = (v4f*)(out_samples + (size_t)ob * DIM);
#pragma unroll
        for (int q = 0; q < 4; ++q) dst[q] = src[q];
    }
}

extern "C" void kernel_launch(void* const* d_in, const int* in_sizes, int n_in,
                              void* d_out, int out_size, void* d_ws, size_t ws_size,
                              hipStream_t stream) {
    const float* mu_q   = (const float*)d_in[0];   // [bs,16]
    const float* std_q  = (const float*)d_in[1];   // [bs,16]
    const float* prior  = (const float*)d_in[2];   // [n_samples,16]
    // d_in[3] (normal_log_prob) is analytically folded into the p^2 coefficient.

    const int bs        = in_sizes[0] / DIM;       // 1024
    const int n_samples = in_sizes[2] / DIM;       // 4096

    float* out_samples = (float*)d_out;                            // [bs,16]
    int*   out_idx     = (int*)((float*)d_out + (size_t)bs * DIM); // [bs]

    dim3 grid(bs / BROWS);    // 64 blocks
    dim3 block(32 * WAVES);   // 8 wave32 waves
    gpfr_score_argmax_kernel<<<grid, block, 0, stream>>>(
        mu_q, std_q, prior, out_samples, out_idx, n_samples);
}